// HyperAttn_ne_86998857548373
// MI455X (gfx1250) — compile-verified
//
#include <hip/hip_runtime.h>
#include <math.h>

typedef float v2f __attribute__((ext_vector_type(2)));
typedef float v8f __attribute__((ext_vector_type(8)));

// ---------------------------------------------------------------------------
// GEMM: C[M,N] = A[M,128] @ B[128,N] + bias[N], fp32 via V_WMMA_F32_16X16X4_F32.
// N is a compile-time constant -> all A/B/C offsets are load/store immediates.
// One wave computes a 16 x (16*TPW) output strip: one A fragment load feeds
// TPW WMMAs (A reuse). Out-of-range columns (N=40) use clamped B addresses:
// B column n only affects D column n, which is never stored, so no zeroing
// and no divergence is needed.
// ---------------------------------------------------------------------------
template <int N>
__global__ __launch_bounds__(256) void wmma_gemm_f32(
    const float* __restrict__ A, const float* __restrict__ B,
    const float* __restrict__ bias, float* __restrict__ C, int M) {
  constexpr int K   = 128;
  constexpr int NTN = (N + 15) / 16;          // 16-wide column tiles
  constexpr int TPW = (NTN >= 4) ? 4 : NTN;   // column tiles per wave
  constexpr int WPM = NTN / TPW;              // waves per 16-row strip
  static_assert(NTN % TPW == 0, "column tiles must split evenly");

  int wave = (int)((blockIdx.x * blockDim.x + threadIdx.x) >> 5);
  int lane = (int)(threadIdx.x & 31);
  int tiles_m = M >> 4;
  if (wave >= tiles_m * WPM) return;  // wave-uniform: EXEC all-ones for WMMA

  int tm = wave / WPM;
  int tw = wave - tm * WPM;
  int m0 = tm << 4;
  int n0 = tw * (TPW * 16);

  int hi   = lane >> 4;   // 0: lanes 0-15, 1: lanes 16-31
  int l15  = lane & 15;
  int koff = hi * 2;      // A/B fragment K sub-offset per lane half

  const float* Arow = A + (size_t)(m0 + l15) * K + koff;

  // Per-column-tile B pointers (clamped for the ragged N=40 tail).
  const float* Bp[TPW];
  int bcol[TPW];
#pragma unroll
  for (int t = 0; t < TPW; ++t) {
    bcol[t] = n0 + t * 16 + l15;
    int cc = (bcol[t] < N) ? bcol[t] : (N - 1);
    Bp[t] = B + (size_t)koff * N + cc;
  }

  v8f acc[TPW];
#pragma unroll
  for (int t = 0; t < TPW; ++t) acc[t] = (v8f){};

#pragma unroll
  for (int k0 = 0; k0 < K; k0 += 4) {
    // A 16x4 fragment: lane holds A[m0+l15, k0+koff .. +1] (8B-aligned pair)
    v2f a = *(const v2f*)(Arow + k0);
#pragma unroll
    for (int t = 0; t < TPW; ++t) {
      v2f b;
      b.x = Bp[t][(size_t)k0 * N];        // B[k0+koff,   col] (imm offset)
      b.y = Bp[t][(size_t)(k0 + 1) * N];  // B[k0+koff+1, col] (imm offset)
      acc[t] = __builtin_amdgcn_wmma_f32_16x16x4_f32(
          /*neg_a=*/false, a, /*neg_b=*/false, b,
          /*c_mod=*/(short)0, acc[t], /*reuse_a=*/false, /*reuse_b=*/false);
    }
  }

  int rbase = m0 + hi * 8;  // C/D layout: VGPR r = rows r and r+8
#pragma unroll
  for (int t = 0; t < TPW; ++t) {
    if (bcol[t] < N) {
      float bv = bias[bcol[t]];
      float* Cp = C + (size_t)rbase * N + bcol[t];
#pragma unroll
      for (int r = 0; r < 8; ++r)
        Cp[(size_t)r * N] = acc[t][r] + bv;
    }
  }
}

template <int N>
static inline void launch_gemm(const float* A, const float* B, const float* bias,
                               float* C, int M, hipStream_t stream) {
  constexpr int NTN = (N + 15) / 16;
  constexpr int TPW = (NTN >= 4) ? 4 : NTN;
  constexpr int WPM = NTN / TPW;
  long long waves = (long long)(M / 16) * WPM;
  int blocks = (int)((waves + 7) / 8);  // 8 waves / 256-thread block
  hipLaunchKernelGGL((wmma_gemm_f32<N>), dim3(blocks), dim3(256), 0, stream,
                     A, B, bias, C, M);
}

// ---------------------------------------------------------------------------
// Monotone float <-> sortable-uint mapping for atomic segment-max.
// Sentinel 0u (empty segment) decodes to NaN -> clamped to 0, matching the
// reference's where(isfinite(m), m, 0).
// ---------------------------------------------------------------------------
__device__ __forceinline__ unsigned f32_order(float f) {
  unsigned b = __float_as_uint(f);
  return (b & 0x80000000u) ? ~b : (b | 0x80000000u);
}
__device__ __forceinline__ float f32_unorder(unsigned u) {
  unsigned b = (u & 0x80000000u) ? (u & 0x7FFFFFFFu) : ~u;
  return __uint_as_float(b);
}

// s[i] = leaky_relu(dot64(kvec[ik[i]], qvec[iq[i]])) * 0.125 ; seg-max via u32
__global__ __launch_bounds__(256) void score_max_kernel(
    const float* __restrict__ kvec, const float* __restrict__ qvec,
    const int* __restrict__ ik, const int* __restrict__ iq,
    const int* __restrict__ seg, float* __restrict__ s_out,
    unsigned* __restrict__ seg_max, int n_inc) {
  int i = (int)(blockIdx.x * blockDim.x + threadIdx.x);
  if (i >= n_inc) return;
  const float4* k4 = (const float4*)(kvec + (size_t)ik[i] * 64);
  const float4* q4 = (const float4*)(qvec + (size_t)iq[i] * 64);
  float acc = 0.0f;
#pragma unroll
  for (int j = 0; j < 16; ++j) {
    float4 kk = k4[j], qq = q4[j];
    acc += kk.x * qq.x + kk.y * qq.y + kk.z * qq.z + kk.w * qq.w;
  }
  float s = (acc > 0.0f ? acc : 0.01f * acc) * 0.125f;  // leaky_relu * 1/sqrt(64)
  s_out[i] = s;
  atomicMax(seg_max + seg[i], f32_order(s));
}

// e[i] = exp(s[i] - m[seg]); denom[seg] += e  (e overwrites s in place)
__global__ __launch_bounds__(256) void exp_denom_kernel(
    float* __restrict__ s_buf, const int* __restrict__ seg,
    const unsigned* __restrict__ seg_max, float* __restrict__ denom,
    int n_inc) {
  int i = (int)(blockIdx.x * blockDim.x + threadIdx.x);
  if (i >= n_inc) return;
  int sg = seg[i];
  float m = f32_unorder(seg_max[sg]);
  if (!__builtin_isfinite(m)) m = 0.0f;
  float e = expf(s_buf[i] - m);
  s_buf[i] = e;
  atomicAdd(denom + sg, e);
}

// out[seg[i], :128] += (e[i]/denom[seg[i]]) * vrows[iv[i], :128]
// One wave per edge: 32 lanes x float4 = 128 columns.
__global__ __launch_bounds__(256) void scatter_kernel(
    const float* __restrict__ e_buf, const float* __restrict__ denom,
    const float* __restrict__ vrows, const int* __restrict__ iv,
    const int* __restrict__ seg, float* __restrict__ out, int n_inc) {
  int gid  = (int)(blockIdx.x * blockDim.x + threadIdx.x);
  int wave = gid >> 5;
  int lane = gid & 31;
  if (wave >= n_inc) return;
  int sg = seg[wave];
  float a = e_buf[wave] / denom[sg];
  const float4* v4 = (const float4*)(vrows + (size_t)iv[wave] * 128);
  float4 v = v4[lane];
  float* dst = out + (size_t)sg * 128 + lane * 4;
  atomicAdd(dst + 0, a * v.x);
  atomicAdd(dst + 1, a * v.y);
  atomicAdd(dst + 2, a * v.z);
  atomicAdd(dst + 3, a * v.w);
}

// ---------------------------------------------------------------------------
extern "C" void kernel_launch(void* const* d_in, const int* in_sizes, int n_in,
                              void* d_out, int out_size, void* d_ws, size_t ws_size,
                              hipStream_t stream) {
  (void)n_in; (void)out_size; (void)ws_size;

  const float* vfeat  = (const float*)d_in[0];
  const float* efeat  = (const float*)d_in[1];
  const int*   inc_n  = (const int*)d_in[2];
  const int*   inc_h  = (const int*)d_in[3];
  const float* W_vtx1 = (const float*)d_in[4];
  const float* b_vtx1 = (const float*)d_in[5];
  const float* W_qe   = (const float*)d_in[6];
  const float* b_qe   = (const float*)d_in[7];
  const float* W_kv   = (const float*)d_in[8];
  const float* b_kv   = (const float*)d_in[9];
  const float* W_vv   = (const float*)d_in[10];
  const float* b_vv   = (const float*)d_in[11];
  const float* W_qv   = (const float*)d_in[12];
  const float* b_qv   = (const float*)d_in[13];
  const float* W_ke   = (const float*)d_in[14];
  const float* b_ke   = (const float*)d_in[15];
  const float* W_ve   = (const float*)d_in[16];
  const float* b_ve   = (const float*)d_in[17];
  const float* W_cls  = (const float*)d_in[18];
  const float* b_cls  = (const float*)d_in[19];

  const int NV = in_sizes[0] / 128;  // 100000
  const int NH = in_sizes[1] / 128;  // 20000
  const int NE = in_sizes[2];        // 1000000

  // Output layout: feat_v_out [NV,128] | feat_e [NH,128] | pred [NV,40]
  float* feat_v_out = (float*)d_out;
  float* feat_e     = feat_v_out + (size_t)NV * 128;
  float* pred       = feat_e + (size_t)NH * 128;

  // Workspace layout (floats)
  float* ws     = (float*)d_ws;
  float* feat_v = ws;                              size_t o = (size_t)NV * 128;
  float* k_v    = ws + o;                          o += (size_t)NV * 64;
  float* v_v    = ws + o;                          o += (size_t)NV * 128;
  float* q_e    = ws + o;                          o += (size_t)NH * 64;
  float* q_v    = ws + o;                          o += (size_t)NV * 64;
  float* k_e    = ws + o;                          o += (size_t)NH * 64;
  float* v_e    = ws + o;                          o += (size_t)NH * 128;
  float* s_buf  = ws + o;                          o += (size_t)NE;
  unsigned* smax1 = (unsigned*)(ws + o);           o += (size_t)NH;
  float*    den1  = ws + o;                        o += (size_t)NH;
  unsigned* smax2 = (unsigned*)(ws + o);           o += (size_t)NV;
  float*    den2  = ws + o;                        o += (size_t)NV;

  // Zero the accumulated outputs and softmax statistics (capture-safe).
  hipMemsetAsync(feat_v_out, 0, (size_t)(NV + NH) * 128 * sizeof(float), stream);
  hipMemsetAsync(smax1, 0, ((size_t)NH * 2 + (size_t)NV * 2) * sizeof(float), stream);

  const int TPB = 256;
  int eb  = (NE + TPB - 1) / TPB;        // 1 thread / edge
  int ewb = (NE * 32 + TPB - 1) / TPB;   // 1 wave / edge

  // Dense projections (fp32 WMMA)
  launch_gemm<128>(vfeat,  W_vtx1, b_vtx1, feat_v, NV, stream);
  launch_gemm<64 >(feat_v, W_kv,   b_kv,   k_v,    NV, stream);
  launch_gemm<128>(feat_v, W_vv,   b_vv,   v_v,    NV, stream);
  launch_gemm<64 >(efeat,  W_qe,   b_qe,   q_e,    NH, stream);

  // Stage 1: nodes -> hyperedges, segmented by inc_hedge
  hipLaunchKernelGGL(score_max_kernel, dim3(eb), dim3(TPB), 0, stream,
                     k_v, q_e, inc_n, inc_h, inc_h, s_buf, smax1, NE);
  hipLaunchKernelGGL(exp_denom_kernel, dim3(eb), dim3(TPB), 0, stream,
                     s_buf, inc_h, smax1, den1, NE);
  hipLaunchKernelGGL(scatter_kernel, dim3(ewb), dim3(TPB), 0, stream,
                     s_buf, den1, v_v, inc_n, inc_h, feat_e, NE);

  // Stage 2 projections
  launch_gemm<64 >(feat_e, W_ke, b_ke, k_e, NH, stream);
  launch_gemm<128>(feat_e, W_ve, b_ve, v_e, NH, stream);
  launch_gemm<64 >(feat_v, W_qv, b_qv, q_v, NV, stream);

  // Stage 2: hyperedges -> nodes, segmented by inc_node
  hipLaunchKernelGGL(score_max_kernel, dim3(eb), dim3(TPB), 0, stream,
                     k_e, q_v, inc_h, inc_n, inc_n, s_buf, smax2, NE);
  hipLaunchKernelGGL(exp_denom_kernel, dim3(eb), dim3(TPB), 0, stream,
                     s_buf, inc_n, smax2, den2, NE);
  hipLaunchKernelGGL(scatter_kernel, dim3(ewb), dim3(TPB), 0, stream,
                     s_buf, den2, v_e, inc_h, inc_n, feat_v_out, NE);

  // Classifier head (N=40, column-predicated stores, clamped B addresses)
  launch_gemm<40 >(feat_v_out, W_cls, b_cls, pred, NV, stream);
}